// GELU56_17566416240653
// MI455X (gfx1250) — compile-verified
//
#include <hip/hip_runtime.h>
#include <hip/hip_bf16.h>
#include <math.h>

// Problem shape (fixed by the reference).
namespace {
constexpr int kB  = 16;
constexpr int kT  = 2048;
constexpr int kD  = 1024;
constexpr int kC  = 128;           // channels (d) handled per block == blockDim.x
constexpr int kTT = 32;            // time steps per LDS tile
constexpr int kNT = kT / kTT;      // 64 tiles
constexpr int kChunksPerThread = (kTT * kC / 4) / kC; // = 8 (16B chunks per thread per tile)
}

#if defined(__has_builtin)
#if __has_builtin(__builtin_amdgcn_global_load_async_to_lds_b128) && \
    __has_builtin(__builtin_amdgcn_s_wait_asynccnt)
#define GELU56_ASYNC_LDS 1
#endif
#endif
#ifndef GELU56_ASYNC_LDS
#define GELU56_ASYNC_LDS 0
#endif

#if GELU56_ASYNC_LDS
// Builtin signature (from hipcc diagnostic): param0 = int __vector(4) __device__*  (AS1),
// param1 expected analogous in LDS space (AS3); offset/cpol are immediates.
typedef int v4i_t __attribute__((vector_size(16)));
typedef __attribute__((address_space(1))) v4i_t* g_v4i_ptr;
typedef __attribute__((address_space(3))) v4i_t* l_v4i_ptr;
#endif

__device__ __forceinline__ float gelu_tanh_approx(float x) {
    const float kC0 = 0.7978845608028654f;            // sqrt(2/pi)
    float u = kC0 * fmaf(0.044715f * x, x * x, x);    // c*(x + 0.044715 x^3)
    // tanh(u) = 1 - 2/(exp(2u)+1); saturates correctly at +/-inf
    float e  = __expf(2.0f * u);
    float th = 1.0f - 2.0f * __builtin_amdgcn_rcpf(e + 1.0f);
    return 0.5f * x * (1.0f + th);
}

// Stage one kTT x kC tile (global row stride kD floats) into lbuf (row-major kC floats/row).
__device__ __forceinline__ void stage_tile(const float* __restrict__ gbase,
                                           float* __restrict__ lbuf, int tid) {
#pragma unroll
    for (int j = 0; j < kChunksPerThread; ++j) {
        int n   = j * kC + tid;       // 16B chunk id within tile
        int row = n >> 5;             // kC/4 = 32 chunks per row
        int col = n & 31;
        const float* g = gbase + (size_t)row * kD + col * 4;
        float*       l = lbuf + n * 4;
#if GELU56_ASYNC_LDS
        __builtin_amdgcn_global_load_async_to_lds_b128(
            (g_v4i_ptr)(g), (l_v4i_ptr)(l), 0, 0);
#else
        const float4 v = *reinterpret_cast<const float4*>(g);
        *reinterpret_cast<float4*>(l) = v;
#endif
    }
}

__global__ __launch_bounds__(kC) void GELU56_17566416240653_kernel(
    const float* __restrict__ x,
    const float* __restrict__ logit_U,
    const float* __restrict__ log_tau_rec,
    const float* __restrict__ ema_resource,
    float* __restrict__ out) {
    __shared__ __align__(16) float smem[2 * kTT * kC];

    const int tid = threadIdx.x;
    const int blk = blockIdx.x;
    const int b   = blk / (kD / kC);
    const int dg  = blk % (kD / kC);
    const int d   = dg * kC + tid;

    // Uniform scalar parameters (scalar-cached loads).
    const float U  = 1.0f / (1.0f + __expf(-logit_U[0]));
    const float sp = log1pf(__expf(log_tau_rec[0]));     // softplus
    const float tau = fmaxf(sp, 0.5f);
    const float rr  = 1.0f - __expf(-1.0f / tau);        // recovery rate

    float r = ema_resource[d];                           // r0 (per-channel)

    const float* gchan = x + ((size_t)b * kT) * kD + (size_t)dg * kC; // tile k=0 base
    const size_t obase = (size_t)b * kT * kD + (size_t)d;

    // Prologue: stage tile 0 into buffer 0.
    stage_tile(gchan, &smem[0], tid);

    for (int k = 0; k < kNT; ++k) {
        if (k + 1 < kNT) {
            stage_tile(gchan + (size_t)(k + 1) * kTT * kD,
                       &smem[((k + 1) & 1) * kTT * kC], tid);
#if GELU56_ASYNC_LDS
            // New tile's 8 asyncs outstanding; retire tile k's (in-order).
            __builtin_amdgcn_s_wait_asynccnt(kChunksPerThread);
#endif
        } else {
#if GELU56_ASYNC_LDS
            __builtin_amdgcn_s_wait_asynccnt(0);
#endif
        }
        __syncthreads();   // tile k visible to all waves in the block

        const float* cur = &smem[(k & 1) * kTT * kC];
        const size_t tglob = obase + (size_t)k * kTT * kD;
#pragma unroll 8
        for (int tl = 0; tl < kTT; ++tl) {
            float v    = cur[tl * kC + tid];
            float g    = gelu_tanh_approx(v);
            float gate = r;                               // PRE-update resource
            float f    = fabsf(g);
            float used = fminf(U * r * f, r * 0.99f);
            r = r - used + (1.0f - r) * rr;
            r = fminf(fmaxf(r, 0.01f), 1.0f);
            out[tglob + (size_t)tl * kD] = g * gate;
        }
        __syncthreads();   // all waves done with buf[k&1] before iter k+1 overwrites it
    }
}

extern "C" void kernel_launch(void* const* d_in, const int* in_sizes, int n_in,
                              void* d_out, int out_size, void* d_ws, size_t ws_size,
                              hipStream_t stream) {
    (void)in_sizes; (void)n_in; (void)out_size; (void)d_ws; (void)ws_size;
    const float* x            = (const float*)d_in[0];
    const float* logit_U      = (const float*)d_in[1];
    const float* log_tau_rec  = (const float*)d_in[2];
    // d_in[3] = logit_decay: does not affect the returned output.
    const float* ema_resource = (const float*)d_in[4];
    float* out = (float*)d_out;

    dim3 grid(kB * (kD / kC));   // 128 blocks
    dim3 block(kC);              // 128 threads = 4 wave32 waves
    hipLaunchKernelGGL(GELU56_17566416240653_kernel, grid, block, 0, stream,
                       x, logit_U, log_tau_rec, ema_resource, out);
}